// SDGNN_New_40475771797813
// MI455X (gfx1250) — compile-verified
//
#include <hip/hip_runtime.h>

// SDGNN: layer1 (community mean->max->relu) + layer2 (1-step KMeans over
// neighbor features with WMMA f32 distance GEMMs), fused gather pipeline.
//
// d_out layout per node n (384 floats):
//   [  0..127] = h1 = relu(concat(x, max_c mean_c))        (written by kernel1)
//   [128..255] = h1  (relu(h1) == h1)                      (written by kernel2)
//   [256..383] = relu(max_c cluster_agg)                   (written by kernel2)

typedef __attribute__((ext_vector_type(2))) float v2f;
typedef __attribute__((ext_vector_type(4))) float v4f;
typedef __attribute__((ext_vector_type(8))) float v8f;

#define DIN   64
#define D2    128
#define NC    4
#define NS    8
#define NM    16
#define OUTD  384
#define WAVES_PER_BLOCK 4

__device__ __forceinline__ void wait_asynccnt0() {
#if __has_builtin(__builtin_amdgcn_s_wait_asynccnt)
  __builtin_amdgcn_s_wait_asynccnt(0);
#else
  asm volatile("s_wait_asynccnt 0" ::: "memory");
#endif
}

// CDNA5 async gather: global (per-lane 16B) -> LDS, no VGPR landing, ASYNCcnt.
__device__ __forceinline__ void async_gather_b128(const float* gsrc, void* lds_dst) {
  unsigned ldsoff = (unsigned)(uintptr_t)lds_dst;   // generic LDS ptr: low 32b = LDS addr
  asm volatile("global_load_async_to_lds_b128 %0, %1, off"
               :: "v"(ldsoff), "v"(gsrc) : "memory");
}

// ---------------------------------------------------------------------------
// Kernel 1: h1 = relu(concat(x, max_c mean_s x[comm]))  -> out[:, 0:128]
// One wave per node; lane handles columns {2*lane, 2*lane+1}.
// ---------------------------------------------------------------------------
__global__ __launch_bounds__(128) void sdgnn_layer1(
    const float* __restrict__ feat, const int* __restrict__ comm,
    float* __restrict__ out, int N)
{
  const int lane = threadIdx.x & 31;
  const int wid  = threadIdx.x >> 5;
  int n = blockIdx.x * WAVES_PER_BLOCK + wid;
  if (n >= N) return;

  const float* xr = feat + (size_t)n * DIN;
  v2f x = *(const v2f*)(xr + lane * 2);

  // one coalesced load of all 32 community indices, broadcast via shuffles
  const int cidxv = comm[(size_t)n * (NC * NS) + lane];

  v2f mx;
  for (int c = 0; c < NC; ++c) {
    v2f acc = {0.f, 0.f};
#pragma unroll
    for (int s = 0; s < NS; ++s) {
      int idx = __shfl(cidxv, c * NS + s);
      v2f v = *(const v2f*)(feat + (size_t)idx * DIN + lane * 2);
      acc += v;
    }
    acc *= 0.125f;
    if (c == 0) mx = acc;
    else { mx.x = fmaxf(mx.x, acc.x); mx.y = fmaxf(mx.y, acc.y); }
  }

  float* o = out + (size_t)n * OUTD;
  v2f h1a = { fmaxf(x.x, 0.f),  fmaxf(x.y, 0.f)  };
  v2f h1b = { fmaxf(mx.x, 0.f), fmaxf(mx.y, 0.f) };
  *(v2f*)(o + lane * 2)       = h1a;
  *(v2f*)(o + DIN + lane * 2) = h1b;
}

// ---------------------------------------------------------------------------
// Kernel 2: one wave per node, per-wave LDS scratch.
// ---------------------------------------------------------------------------
struct WaveScratch {
  float neigh[NM][D2];   // 8 KB: gathered neighbor h1 rows (row-major)
  float c0[NC][D2];      // 2 KB: initial centers
  float c1[NC][D2];      // 2 KB: updated centers
  float dotb[NM][NC];    // dot[m][c] spilled from WMMA D layout
  int   labels[NM];
  float cnt[NC];
  float nrmA[NC];
  float nrmB[NC];
};

// Distance-argmin assign step.  dot = neigh[16x128] x ctr^T[128x4] computed as
// 32 chained V_WMMA_F32_16X16X4_F32 ops (exact f32).
// A 16x4 layout (ISA 7.12.2): lane<16 -> row=lane, v0=K0,v1=K1 ; lane>=16 ->
// row=lane-16, v0=K2,v1=K3.  B 4x16 mirrored: lane = output column (cluster);
// columns 4..15 are don't-care (clamped index, outputs ignored).
__device__ __forceinline__ void assign_labels(
    WaveScratch& S, const float (*ctr)[D2], const float* nrm, int lane)
{
  const int mrow = lane & 15;
  const int koff = (lane < 16) ? 0 : 2;
  const int ccol = lane & 3;           // clamp: lanes 4..15 / 20..31 -> garbage cols
  v8f acc = {};
#pragma unroll
  for (int t = 0; t < 32; ++t) {
    v2f a = *(const v2f*)&S.neigh[mrow][4 * t + koff];
    v2f b = *(const v2f*)&ctr[ccol][4 * t + koff];
    acc = __builtin_amdgcn_wmma_f32_16x16x4_f32(
        /*neg_a=*/false, a, /*neg_b=*/false, b,
        /*c_mod=*/(short)0, acc, /*reuse_a=*/false, /*reuse_b=*/false);
  }
  // D layout: vgpr k, lanes 0-15 -> D[M=k][N=lane]; lanes 16-31 -> D[M=k+8].
  if ((lane & 15) < NC) {
    const int c  = lane & 3;
    const int mb = (lane < 16) ? 0 : 8;
#pragma unroll
    for (int k = 0; k < 8; ++k) S.dotb[mb + k][c] = acc[k];
  }
  __syncthreads();

  if (lane < NM) {
    float best = nrm[0] - 2.f * S.dotb[lane][0];
    int bi = 0;
#pragma unroll
    for (int c = 1; c < NC; ++c) {
      float d = nrm[c] - 2.f * S.dotb[lane][c];
      if (d < best) { best = d; bi = c; }   // strict <: first-min like argmin
    }
    S.labels[lane] = bi;
  }
  if (lane < NC) S.cnt[lane] = 0.f;
  __syncthreads();
  if (lane < NM) atomicAdd(&S.cnt[S.labels[lane]], 1.f);
  __syncthreads();
}

__device__ __forceinline__ void cluster_sums(WaveScratch& S, int lane, v4f sum[NC])
{
#pragma unroll
  for (int c = 0; c < NC; ++c) sum[c] = (v4f){0.f, 0.f, 0.f, 0.f};
  for (int m = 0; m < NM; ++m) {
    int lab = S.labels[m];                        // wave-uniform
    v4f v = *(const v4f*)&S.neigh[m][lane * 4];
#pragma unroll
    for (int c = 0; c < NC; ++c) {
      float msk = (lab == c) ? 1.f : 0.f;
      sum[c] += msk * v;
    }
  }
}

__global__ __launch_bounds__(128) void sdgnn_layer2(
    const int* __restrict__ comm, const int* __restrict__ nidx,
    float* __restrict__ out, int N)
{
  __shared__ WaveScratch sm[WAVES_PER_BLOCK];
  const int lane = threadIdx.x & 31;
  const int wid  = threadIdx.x >> 5;
  WaveScratch& S = sm[wid];

  int n = blockIdx.x * WAVES_PER_BLOCK + wid;
  if (n >= N) n = N - 1;   // duplicate (deterministic) work keeps barriers uniform

  // coalesced index loads + shuffle broadcast (wave-uniform values)
  const int nidxv = nidx[(size_t)n * NM + (lane & 15)];
  const int cidxv = comm[(size_t)n * (NC * NS) + lane];

  // ---- issue async gathers: neighbor h1 rows -> LDS directly (ASYNCcnt) ----
#pragma unroll
  for (int m = 0; m < NM; ++m) {
    int r = __shfl(nidxv, m);
    async_gather_b128(out + (size_t)r * OUTD + lane * 4, &S.neigh[m][lane * 4]);
  }

  if (lane < NC) { S.nrmA[lane] = 0.f; S.nrmB[lane] = 0.f; }
  __syncthreads();

  // ---- initial centers = per-community mean of h1 rows, plus ||c||^2 ----
  // (VGPR-accumulated global gathers; overlaps with the async neigh DMA)
  for (int c = 0; c < NC; ++c) {
    v4f a = {0.f, 0.f, 0.f, 0.f};
#pragma unroll
    for (int s = 0; s < NS; ++s) {
      int r = __shfl(cidxv, c * NS + s);
      a += *(const v4f*)(out + (size_t)r * OUTD + lane * 4);
    }
    a *= 0.125f;
    *(v4f*)&S.c0[c][lane * 4] = a;
    atomicAdd(&S.nrmA[c], a.x * a.x + a.y * a.y + a.z * a.z + a.w * a.w);
  }

  wait_asynccnt0();          // neigh rows resident in LDS
  __syncthreads();

  // ---- E-step on init centers ----
  assign_labels(S, S.c0, S.nrmA, lane);

  // ---- M-step: centers1 = cnt>0 ? scatter-mean : c0 ----
  v4f sum[NC];
  cluster_sums(S, lane, sum);
#pragma unroll
  for (int c = 0; c < NC; ++c) {
    float cc = S.cnt[c];
    v4f v;
    if (cc > 0.f) v = sum[c] / cc;
    else          v = *(const v4f*)&S.c0[c][lane * 4];
    *(v4f*)&S.c1[c][lane * 4] = v;
    atomicAdd(&S.nrmB[c], v.x * v.x + v.y * v.y + v.z * v.z + v.w * v.w);
  }
  __syncthreads();

  // ---- final E-step + scatter-mean (empty cluster -> 0) ----
  assign_labels(S, S.c1, S.nrmB, lane);
  cluster_sums(S, lane, sum);

  v4f mx;
#pragma unroll
  for (int c = 0; c < NC; ++c) {
    float cc = S.cnt[c];
    v4f v = {0.f, 0.f, 0.f, 0.f};
    if (cc > 0.f) v = sum[c] / cc;
    if (c == 0) mx = v;
    else {
      mx.x = fmaxf(mx.x, v.x); mx.y = fmaxf(mx.y, v.y);
      mx.z = fmaxf(mx.z, v.z); mx.w = fmaxf(mx.w, v.w);
    }
  }
  mx.x = fmaxf(mx.x, 0.f); mx.y = fmaxf(mx.y, 0.f);
  mx.z = fmaxf(mx.z, 0.f); mx.w = fmaxf(mx.w, 0.f);

  // ---- outputs: [128..255]=h1, [256..383]=relu(max agg) ----
  float* o = out + (size_t)n * OUTD;
  v4f h1v = *(const v4f*)(o + lane * 4);
  *(v4f*)(o + D2 + lane * 4)     = h1v;
  *(v4f*)(o + 2 * D2 + lane * 4) = mx;
}

extern "C" void kernel_launch(void* const* d_in, const int* in_sizes, int n_in,
                              void* d_out, int out_size, void* d_ws, size_t ws_size,
                              hipStream_t stream) {
  const float* feat = (const float*)d_in[0];
  const int*   comm = (const int*)d_in[1];
  const int*   nidx = (const int*)d_in[2];
  float*       out  = (float*)d_out;

  const int N = in_sizes[0] / DIN;
  const int blocks = (N + WAVES_PER_BLOCK - 1) / WAVES_PER_BLOCK;

  sdgnn_layer1<<<blocks, 128, 0, stream>>>(feat, comm, out, N);
  sdgnn_layer2<<<blocks, 128, 0, stream>>>(comm, nidx, out, N);
}